// SparseDepthLabeler_18133351923970
// MI455X (gfx1250) — compile-verified
//
#include <hip/hip_runtime.h>
#include <hip/hip_bf16.h>

typedef float v2f __attribute__((ext_vector_type(2)));
typedef float v8f __attribute__((ext_vector_type(8)));

// ---------------------------------------------------------------------------
// Kernel 1: init z-buffer (lives in d_out) to +inf bit pattern.
// ---------------------------------------------------------------------------
__global__ void SDL_zbuf_init(unsigned int* __restrict__ zbuf, int n) {
    int i = blockIdx.x * blockDim.x + threadIdx.x;
    if (i < n) zbuf[i] = 0x7F800000u;  // +inf
}

// ---------------------------------------------------------------------------
// Kernel 2: project points into all 6 cameras via V_WMMA_F32_16X16X4_F32 and
// z-buffer with atomicMin on positive-float bit patterns.
//
// Per 16-point group:
//   B (4x16) = homogeneous points: lane n    -> (x,y) in V0,V1 (K=0,1)
//                                  lane n+16 -> (z,1) in V0,V1 (K=2,3)
//   A1 (16x4) rows 0-5 = ego2cam rows of cams 0,1 ; rows 8-13 = cams 2,3
//   A2 (16x4) rows 0-5 = ego2cam rows of cams 4,5 ; rest zero
//   D: lane j holds rows 0-7 (v[0..7]), lane j+16 holds rows 8-15.
//   => lane j:    cams 0,1 (W1) + cams 4,5 (W2) of point j
//      lane j+16: cams 2,3 (W1)               of point j
// ---------------------------------------------------------------------------
__global__ __launch_bounds__(256)
void SDL_project(const float* __restrict__ pts,     // (B, Npts, 4)
                 const float* __restrict__ intr,    // (B, 6, 3, 3)
                 const float* __restrict__ c2e,     // (B, 6, 4, 4) cam2ego
                 const int*   __restrict__ pHf,
                 const int*   __restrict__ pWf,
                 unsigned int* __restrict__ zbuf,   // (B, 6, Hf, Wf)
                 int Npts, int wavesPerB) {
    const int Hf = *pHf;
    const int Wf = *pWf;
    const int HW = Hf * Wf;
    const int b  = blockIdx.y;
    const int lane    = threadIdx.x & 31;
    const int m       = lane & 15;          // A-matrix row index held by this lane
    const bool hiHalf = lane >= 16;         // holds K=2,3 of A and B

    // ---- Build A operands (ego2cam = [R^T | -R^T t] rows) once per wave ----
    float a1x = 0.0f, a1y = 0.0f;   // WMMA #1 A: (k0,k1) or (k2,k3)
    float a2x = 0.0f, a2y = 0.0f;   // WMMA #2 A
    {
        int cam1 = -1, cmp1 = 0;
        if (m < 6)                 { cam1 = m / 3;            cmp1 = m % 3; }
        else if (m >= 8 && m < 14) { cam1 = 2 + (m - 8) / 3;  cmp1 = (m - 8) % 3; }
        if (cam1 >= 0) {
            const float* M = c2e + (size_t)(b * 6 + cam1) * 16;
            if (!hiHalf) { a1x = M[0 * 4 + cmp1]; a1y = M[1 * 4 + cmp1]; }
            else {
                a1x = M[2 * 4 + cmp1];
                a1y = -(M[0 * 4 + cmp1] * M[0 * 4 + 3] +
                        M[1 * 4 + cmp1] * M[1 * 4 + 3] +
                        M[2 * 4 + cmp1] * M[2 * 4 + 3]);
            }
        }
        if (m < 6) {
            int cam2 = 4 + m / 3, cmp2 = m % 3;
            const float* M = c2e + (size_t)(b * 6 + cam2) * 16;
            if (!hiHalf) { a2x = M[0 * 4 + cmp2]; a2y = M[1 * 4 + cmp2]; }
            else {
                a2x = M[2 * 4 + cmp2];
                a2y = -(M[0 * 4 + cmp2] * M[0 * 4 + 3] +
                        M[1 * 4 + cmp2] * M[1 * 4 + 3] +
                        M[2 * 4 + cmp2] * M[2 * 4 + 3]);
            }
        }
    }
    v2f A1; A1.x = a1x; A1.y = a1y;
    v2f A2; A2.x = a2x; A2.y = a2y;

    // ---- Per-lane camera parameter preload ----
    // lane<16 handles cams {0,1,4,5}; lane>=16 handles cams {2,3}
    int myCams[4];
    myCams[0] = hiHalf ? 2 : 0;
    myCams[1] = myCams[0] + 1;
    myCams[2] = 4;
    myCams[3] = 5;
    float FX[4], FY[4], CX[4], CY[4];
    int   SEG[4];
    #pragma unroll
    for (int q = 0; q < 4; ++q) {
        const float* Kp = intr + (size_t)(b * 6 + myCams[q]) * 9;
        FX[q] = Kp[0]; FY[q] = Kp[4]; CX[q] = Kp[2]; CY[q] = Kp[5];
        SEG[q] = (b * 6 + myCams[q]) * HW;
    }

    auto proj = [&](float x, float y, float z, int q) {
        float zc  = fmaxf(z, 1e-6f);
        // Hardware V_RCP_F32 (1 ulp): error ~1e-4 px vs 4-px bins -> safe here,
        // and ~10x fewer VALU ops than the IEEE div expansion in the hot loop.
        float inv = __builtin_amdgcn_rcpf(zc);
        float u = fmaf(FX[q], x * inv, CX[q]);
        float v = fmaf(FY[q], y * inv, CY[q]);
        bool ok = (z > 0.1f) && (u >= 0.0f) && (u <= 703.0f)
                             && (v >= 0.0f) && (v <= 255.0f);
        if (ok) {
            int uf = (int)floorf(u * 0.25f);
            int vf = (int)floorf(v * 0.25f);
            if (uf >= 0 && uf < Wf && vf >= 0 && vf < Hf) {
                atomicMin(&zbuf[SEG[q] + vf * Wf + uf], __float_as_uint(z));
            }
        }
    };

    // ---- Main loop over 16-point groups ----
    const float* ptsB = pts + (size_t)b * Npts * 4;
    const int groups = (Npts + 15) >> 4;
    const int waveInB = blockIdx.x * (blockDim.x >> 5) + (threadIdx.x >> 5);
    const int bOff = hiHalf ? 2 : 0;   // lo half reads (x,y); hi half reads (z, .)

    for (int g = waveInB; g < groups; g += wavesPerB) {
        int p  = (g << 4) + m;
        int pc = min(p, Npts - 1);
        // prefetch next group's data for this lane
        int gn = g + wavesPerB;
        if (gn < groups) {
            __builtin_prefetch(ptsB + (size_t)((gn << 4) + m) * 4 + bOff, 0, 1);
        }
        v2f pv = *(const v2f*)(ptsB + (size_t)pc * 4 + bOff);
        float bx = pv.x;
        float by = hiHalf ? 1.0f : pv.y;          // homogeneous w = 1
        if (p >= Npts) { bx = 0.0f; by = 0.0f; }  // tail -> z_cam=0 -> invalid
        v2f Bv; Bv.x = bx; Bv.y = by;

        v8f c0 = {};
        v8f d1 = __builtin_amdgcn_wmma_f32_16x16x4_f32(
            false, A1, false, Bv, (short)0, c0, false, false);
        v8f d2 = __builtin_amdgcn_wmma_f32_16x16x4_f32(
            false, A2, false, Bv, (short)0, c0, false, false);

        proj(d1[0], d1[1], d1[2], 0);   // cam 0 (lo) / cam 2 (hi)
        proj(d1[3], d1[4], d1[5], 1);   // cam 1 (lo) / cam 3 (hi)
        if (!hiHalf) {
            proj(d2[0], d2[1], d2[2], 2);   // cam 4
            proj(d2[3], d2[4], d2[5], 3);   // cam 5
        }
    }
}

// ---------------------------------------------------------------------------
// Kernel 3: in-place convert min-depth bits -> int32 bin labels.
// ---------------------------------------------------------------------------
__global__ void SDL_finalize(unsigned int* __restrict__ zbuf_io,
                             const float* __restrict__ edges,
                             int nEdges, int n) {
    int i = blockIdx.x * blockDim.x + threadIdx.x;
    if (i >= n) return;
    unsigned int bits = zbuf_io[i];
    int label = -1;
    if (bits != 0x7F800000u) {                 // finite min depth exists
        float z  = __uint_as_float(bits);
        float lo = edges[0] + 0.001f;
        float hi = edges[nEdges - 1] - 0.001f;
        float d  = fminf(fmaxf(z, lo), hi);
        // searchsorted(edges, d, side='left') - 1
        int idx = nEdges;
        for (int k = 0; k < nEdges; ++k) {
            if (edges[k] >= d) { idx = k; break; }
        }
        int bin = idx - 1;
        bin = max(0, min(bin, nEdges - 2));
        label = bin;
    }
    ((int*)zbuf_io)[i] = label;
}

// ---------------------------------------------------------------------------
extern "C" void kernel_launch(void* const* d_in, const int* in_sizes, int n_in,
                              void* d_out, int out_size, void* d_ws, size_t ws_size,
                              hipStream_t stream) {
    (void)n_in; (void)d_ws; (void)ws_size;
    const float* pts   = (const float*)d_in[0];   // (B, Npts, 4)
    const float* intr  = (const float*)d_in[1];   // (B, 6, 3, 3)
    const float* c2e   = (const float*)d_in[2];   // (B, 6, 4, 4)
    const float* edges = (const float*)d_in[3];   // (49,)
    const int*   pHf   = (const int*)d_in[4];
    const int*   pWf   = (const int*)d_in[5];
    unsigned int* zbuf = (unsigned int*)d_out;

    const int Ncam   = 6;                               // per setup_inputs
    const int B      = (in_sizes[1] / 9) / Ncam;        // intrinsics = B*6*9
    const int Npts   = in_sizes[0] / (B * 4);
    const int nEdges = in_sizes[3];
    const int n      = out_size;

    dim3 blk(256);
    SDL_zbuf_init<<<dim3((n + 255) / 256), blk, 0, stream>>>(zbuf, n);

    const int blocksPerB = 512;                 // 512 blocks * 8 waves = 4096 waves/b
    const int wavesPerB  = blocksPerB * (256 / 32);
    dim3 grid(blocksPerB, B);
    SDL_project<<<grid, blk, 0, stream>>>(pts, intr, c2e, pHf, pWf, zbuf,
                                          Npts, wavesPerB);

    SDL_finalize<<<dim3((n + 255) / 256), blk, 0, stream>>>(zbuf, edges, nEdges, n);
}